// BDH_GPU_412316861083
// MI455X (gfx1250) — compile-verified
//
#include <hip/hip_runtime.h>

// BDH recurrence reformulated as linear attention:
//   x_t  : cheap sequential scan (decay + L1 norm), parallel over B*n
//   a*_t = sum_{s<t} 0.97^{t-1-s} (x_s . x_t) ln(v_s)   -> Gram GEMM + masked GEMM
//   everything else: big parallel GEMMs (WMMA f32 16x16x4) + layernorms.
// Working set ~36MB -> fully L2-resident on MI455X (192MB L2).
// GEMM waves are register-blocked 16x64 (4 accumulators) so each A-fragment
// load feeds 4 v_wmma ops.

typedef __attribute__((ext_vector_type(2))) float v2f;
typedef __attribute__((ext_vector_type(8))) float v8f;

static constexpr int Bb = 8;
static constexpr int Tt = 256;
static constexpr int Dd = 256;
static constexpr int Nn = 1024;

// ---------------- LayerNorm over last dim (256), no affine ----------------
__global__ __launch_bounds__(256) void ln256_kernel(const float* __restrict__ in,
                                                    float* __restrict__ out) {
  const int D = 256;
  const long row = blockIdx.x;
  const int j = threadIdx.x;
  float v = in[row * D + j];
  __shared__ float red[8];

  float s = v;
#pragma unroll
  for (int o = 16; o > 0; o >>= 1) s += __shfl_down(s, o, 32);
  if ((j & 31) == 0) red[j >> 5] = s;
  __syncthreads();
  float tot = 0.f;
#pragma unroll
  for (int w = 0; w < 8; ++w) tot += red[w];
  const float mean = tot * (1.0f / 256.0f);
  const float c = v - mean;
  __syncthreads();

  float s2 = c * c;
#pragma unroll
  for (int o = 16; o > 0; o >>= 1) s2 += __shfl_down(s2, o, 32);
  if ((j & 31) == 0) red[j >> 5] = s2;
  __syncthreads();
  float tot2 = 0.f;
#pragma unroll
  for (int w = 0; w < 8; ++w) tot2 += red[w];
  const float var = tot2 * (1.0f / 256.0f);

  out[row * D + j] = c * rsqrtf(var + 1e-5f);
}

// ---------------- sequential x-state scan (decay + L1 normalize) ----------------
__global__ __launch_bounds__(1024) void xscan_kernel(const float* __restrict__ XU,
                                                     float* __restrict__ X) {
  const int b = blockIdx.x;
  const int j = threadIdx.x;  // n index, blockDim = 1024 (32 waves)
  const float* xu = XU + (long)b * Tt * Nn;
  float* xo = X + (long)b * Tt * Nn;
  __shared__ float red[32];
  float x = 0.f;
  for (int t = 0; t < Tt; ++t) {
    x = 0.97f * x + xu[(long)t * Nn + j];
    float s = fabsf(x);
#pragma unroll
    for (int o = 16; o > 0; o >>= 1) s += __shfl_down(s, o, 32);
    if ((j & 31) == 0) red[j >> 5] = s;
    __syncthreads();
    if (j < 32) {
      float v = red[j];
#pragma unroll
      for (int o = 16; o > 0; o >>= 1) v += __shfl_down(v, o, 32);
      if (j == 0) red[0] = v;
    }
    __syncthreads();
    x = x / fmaxf(red[0], 1e-12f);
    xo[(long)t * Nn + j] = x;
    __syncthreads();  // red[] reused next iteration
  }
}

// ---------------- causal decay mask applied to Gram matrix ----------------
__global__ __launch_bounds__(256) void mask_kernel(const float* __restrict__ Kg,
                                                   float* __restrict__ Wm) {
  const int s = threadIdx.x;
  const int t = blockIdx.x;
  const int b = blockIdx.y;
  const long idx = (((long)b * Tt) + t) * Tt + s;
  float v = 0.f;
  if (s < t) v = __powf(0.97f, (float)(t - 1 - s)) * Kg[idx];
  Wm[idx] = v;
}

// ---------------- WMMA f32 GEMM: C = A(MxK,rm) @ W(NxK,rm)^T ----------------
// Wave computes a 16x64 strip: 4 accumulators share one A fragment.
// EPI: 0 = plain, 1 = relu(c), 2 = relu(c) * relu(Mul[row,col])
template <int EPI>
__global__ __launch_bounds__(256) void gemm_ABt_wmma(const float* __restrict__ A,
                                                     const float* __restrict__ W,
                                                     float* __restrict__ C,
                                                     const float* __restrict__ Mul,
                                                     int M, int N, int K,
                                                     long sA, long sW, long sC) {
  const int b = blockIdx.y;
  A += (long)b * sA;
  W += (long)b * sW;
  C += (long)b * sC;
  const int lane = threadIdx.x & 31;
  const int wave = threadIdx.x >> 5;
  const int strip = blockIdx.x * (blockDim.x >> 5) + wave;
  const int nstrips_n = N >> 6;  // strips of 64 cols
  const int tm = strip / nstrips_n;
  const int tn = strip - tm * nstrips_n;
  if (tm >= (M >> 4)) return;  // wave-uniform: EXEC stays all-ones for WMMA

  const int half = lane >> 4;  // K sub-block select per ISA f32 A/B layout
  const int l = lane & 15;

  const float2* Arow = (const float2*)(A + (long)(tm * 16 + l) * K);
  const float2* Wrow0 = (const float2*)(W + (long)(tn * 64 + 0 * 16 + l) * K);
  const float2* Wrow1 = (const float2*)(W + (long)(tn * 64 + 1 * 16 + l) * K);
  const float2* Wrow2 = (const float2*)(W + (long)(tn * 64 + 2 * 16 + l) * K);
  const float2* Wrow3 = (const float2*)(W + (long)(tn * 64 + 3 * 16 + l) * K);

  v8f c0 = {}, c1 = {}, c2 = {}, c3 = {};
#pragma unroll 2
  for (int kk = 0; kk < K; kk += 4) {
    const int ki = (kk >> 1) + half;
    const float2 af = Arow[ki];  // A[m][kk+2*half .. +1]
    v2f a; a[0] = af.x; a[1] = af.y;
    float2 wf;
    v2f bb;
    wf = Wrow0[ki]; bb[0] = wf.x; bb[1] = wf.y;
    c0 = __builtin_amdgcn_wmma_f32_16x16x4_f32(false, a, false, bb, (short)0, c0,
                                               false, false);
    wf = Wrow1[ki]; bb[0] = wf.x; bb[1] = wf.y;
    c1 = __builtin_amdgcn_wmma_f32_16x16x4_f32(false, a, false, bb, (short)0, c1,
                                               false, false);
    wf = Wrow2[ki]; bb[0] = wf.x; bb[1] = wf.y;
    c2 = __builtin_amdgcn_wmma_f32_16x16x4_f32(false, a, false, bb, (short)0, c2,
                                               false, false);
    wf = Wrow3[ki]; bb[0] = wf.x; bb[1] = wf.y;
    c3 = __builtin_amdgcn_wmma_f32_16x16x4_f32(false, a, false, bb, (short)0, c3,
                                               false, false);
  }

  v8f acc[4] = {c0, c1, c2, c3};
#pragma unroll
  for (int t = 0; t < 4; ++t) {
    const long col = tn * 64 + t * 16 + l;
#pragma unroll
    for (int r = 0; r < 8; ++r) {
      const long row = tm * 16 + r + 8 * half;
      float v = acc[t][r];
      if (EPI == 1) v = fmaxf(v, 0.f);
      if (EPI == 2) v = fmaxf(v, 0.f) * fmaxf(Mul[row * N + col], 0.f);
      C[row * N + col] = v;
    }
  }
}

// ---------------- WMMA f32 GEMM: C = A(MxK,rm) @ Bm(KxN,rm) ----------------
__global__ __launch_bounds__(256) void gemm_AB_wmma(const float* __restrict__ A,
                                                    const float* __restrict__ Bm,
                                                    float* __restrict__ C,
                                                    int M, int N, int K,
                                                    long sA, long sB, long sC) {
  const int b = blockIdx.y;
  A += (long)b * sA;
  Bm += (long)b * sB;
  C += (long)b * sC;
  const int lane = threadIdx.x & 31;
  const int wave = threadIdx.x >> 5;
  const int strip = blockIdx.x * (blockDim.x >> 5) + wave;
  const int nstrips_n = N >> 6;
  const int tm = strip / nstrips_n;
  const int tn = strip - tm * nstrips_n;
  if (tm >= (M >> 4)) return;

  const int half = lane >> 4;
  const int l = lane & 15;

  const float2* Arow = (const float2*)(A + (long)(tm * 16 + l) * K);
  const float* Bc0 = Bm + (tn * 64 + 0 * 16 + l);
  const float* Bc1 = Bm + (tn * 64 + 1 * 16 + l);
  const float* Bc2 = Bm + (tn * 64 + 2 * 16 + l);
  const float* Bc3 = Bm + (tn * 64 + 3 * 16 + l);

  v8f c0 = {}, c1 = {}, c2 = {}, c3 = {};
#pragma unroll 2
  for (int kk = 0; kk < K; kk += 4) {
    const float2 af = Arow[(kk >> 1) + half];
    const long k0 = kk + 2 * half;
    v2f a; a[0] = af.x; a[1] = af.y;
    v2f bb;
    bb[0] = Bc0[k0 * N]; bb[1] = Bc0[(k0 + 1) * N];
    c0 = __builtin_amdgcn_wmma_f32_16x16x4_f32(false, a, false, bb, (short)0, c0,
                                               false, false);
    bb[0] = Bc1[k0 * N]; bb[1] = Bc1[(k0 + 1) * N];
    c1 = __builtin_amdgcn_wmma_f32_16x16x4_f32(false, a, false, bb, (short)0, c1,
                                               false, false);
    bb[0] = Bc2[k0 * N]; bb[1] = Bc2[(k0 + 1) * N];
    c2 = __builtin_amdgcn_wmma_f32_16x16x4_f32(false, a, false, bb, (short)0, c2,
                                               false, false);
    bb[0] = Bc3[k0 * N]; bb[1] = Bc3[(k0 + 1) * N];
    c3 = __builtin_amdgcn_wmma_f32_16x16x4_f32(false, a, false, bb, (short)0, c3,
                                               false, false);
  }

  v8f acc[4] = {c0, c1, c2, c3};
#pragma unroll
  for (int t = 0; t < 4; ++t) {
    const long col = tn * 64 + t * 16 + l;
#pragma unroll
    for (int r = 0; r < 8; ++r) {
      const long row = tm * 16 + r + 8 * half;
      C[row * N + col] = acc[t][r];
    }
  }
}

extern "C" void kernel_launch(void* const* d_in, const int* in_sizes, int n_in,
                              void* d_out, int out_size, void* d_ws, size_t ws_size,
                              hipStream_t stream) {
  const float* emb = (const float*)d_in[0];  // (B,T,d)
  const float* E   = (const float*)d_in[1];  // (d,n)
  const float* Dx  = (const float*)d_in[2];  // (n,d)
  const float* Dy  = (const float*)d_in[3];  // (n,d)
  float* out = (float*)d_out;                // (B,T,d)

  const long BT = (long)Bb * Tt;           // 2048 rows
  const long szBTD = BT * Dd;              // 524288
  const long szBTN = BT * Nn;              // 2097152
  const long szBTT = (long)Bb * Tt * Tt;   // 524288

  float* ws = (float*)d_ws;                // needs ~36 MB
  float* LNV = ws;                         // (B,T,d)  ln(embeddings)
  float* XU  = LNV + szBTD;                // (B,T,n)  relu(emb @ Dx^T)
  float* X   = XU + szBTN;                 // (B,T,n)  normalized x_t
  float* Kg  = X + szBTN;                  // (B,T,T)  Gram X X^T
  float* Wm  = Kg + szBTT;                 // (B,T,T)  causal decay-masked Gram
  float* As  = Wm + szBTT;                 // (B,T,d)  a_star
  float* LNA = As + szBTD;                 // (B,T,d)  ln(a_star)
  float* Y   = LNA + szBTD;                // (B,T,n)  y_t
  float* V   = Y + szBTN;                  // (B,T,d)  y_t @ E^T

  // strips-of-64-cols grid helper: blocks = (M/16)*(N/64)/8
  // 1. LNV = ln(embeddings)
  ln256_kernel<<<(int)BT, 256, 0, stream>>>(emb, LNV);

  // 2. XU = relu(emb_flat @ Dx^T)   (2048x256)@(256x1024)
  gemm_ABt_wmma<1><<<dim3((2048 / 16) * (1024 / 64) / 8, 1), 256, 0, stream>>>(
      emb, Dx, XU, nullptr, 2048, 1024, 256, 0, 0, 0);

  // 3. sequential x scan: decay + L1 normalize (only true recurrence left)
  xscan_kernel<<<Bb, 1024, 0, stream>>>(XU, X);

  // 4. Kg[b] = X_b @ X_b^T   (256x1024)@(1024x256), batched over B
  gemm_ABt_wmma<0><<<dim3((256 / 16) * (256 / 64) / 8, Bb), 256, 0, stream>>>(
      X, X, Kg, nullptr, 256, 256, 1024, (long)Tt * Nn, (long)Tt * Nn,
      (long)Tt * Tt);

  // 5. causal decay mask: Wm[t,s] = 0.97^(t-1-s) * Kg[t,s] for s<t
  mask_kernel<<<dim3(Tt, Bb), 256, 0, stream>>>(Kg, Wm);

  // 6. As[b] = Wm_b @ LNV_b   (256x256)@(256x256), batched
  gemm_AB_wmma<<<dim3((256 / 16) * (256 / 64) / 8, Bb), 256, 0, stream>>>(
      Wm, LNV, As, 256, 256, 256, (long)Tt * Tt, (long)Tt * Dd, (long)Tt * Dd);

  // 7. LNA = ln(As)
  ln256_kernel<<<(int)BT, 256, 0, stream>>>(As, LNA);

  // 8. Y = relu(LNA @ Dy^T) * relu(X)   (2048x256)@(256x1024), fused epilogue
  gemm_ABt_wmma<2><<<dim3((2048 / 16) * (1024 / 64) / 8, 1), 256, 0, stream>>>(
      LNA, Dy, Y, X, 2048, 1024, 256, 0, 0, 0);

  // 9. V = Y @ E^T   (2048x1024)@(1024x256)
  gemm_ABt_wmma<0><<<dim3((2048 / 16) * (256 / 64) / 8, 1), 256, 0, stream>>>(
      Y, E, V, nullptr, 2048, 256, 1024, 0, 0, 0);

  // 10. out = ln(V)
  ln256_kernel<<<(int)BT, 256, 0, stream>>>(V, out);
}